// SparseSharedTokenCrossAttention_53085795779214
// MI455X (gfx1250) — compile-verified
//
#include <hip/hip_runtime.h>
#include <hip/hip_bf16.h>

typedef float v2f __attribute__((ext_vector_type(2)));
typedef float v4f __attribute__((ext_vector_type(4)));
typedef float v8f __attribute__((ext_vector_type(8)));

#define TILE_M 32
#define TILE_N 128
#define TILE_K 32
#define LDA 36    // TILE_K + 4 pad; rows 144B apart -> 16B aligned, bank-safe
#define LDB 132   // TILE_N + 4 pad; rows 528B apart -> 16B aligned

#define ATT_HEADS 8
#define ATT_DH    64
#define ATT_KN    32
#define ATT_INNER 512
#define ATT_SCALE 0.125f   // 64^-0.5

// Issue one 16B async global->LDS transfer (GV addressing, ASYNCcnt-tracked).
__device__ __forceinline__ void async_copy16(const float* gsrc, const float* ldst) {
    const unsigned lds = (unsigned)(uintptr_t)ldst;                // low 32b = LDS offset
    const unsigned long long g = (unsigned long long)(uintptr_t)gsrc;
    asm volatile("global_load_async_to_lds_b128 %0, %1, off"
                 :: "v"(lds), "v"(g) : "memory");
}
__device__ __forceinline__ void wait_async_all() {
    asm volatile("s_wait_asynccnt 0x0" ::: "memory");
}

// C[M,N] = A[M,K] @ B[K,N] (+ bias[N]); row-major fp32.
// Requires M%32==0, N%128==0, K%32==0 (true for all three GEMMs here).
__global__ __launch_bounds__(256)
void gemm_wmma_f32(const float* __restrict__ A, const float* __restrict__ B,
                   const float* __restrict__ bias, float* __restrict__ C,
                   int M, int N, int K) {
    __shared__ __align__(16) float sA[2][TILE_M * LDA];   // 9.2 KB
    __shared__ __align__(16) float sB[2][TILE_K * LDB];   // 33.8 KB

    const int tid  = threadIdx.x;
    const int wave = tid >> 5;            // 0..7
    const int lane = tid & 31;
    const int wm0  = (wave >> 2) << 4;    // M sub-tile: 0 or 16
    const int wn0  = (wave & 3) << 5;     // N sub-tile: 0,32,64,96 (two 16-wide tiles)
    const int half = lane >> 4;
    const int lrow = lane & 15;

    const int m0 = blockIdx.y * TILE_M;
    const int n0 = blockIdx.x * TILE_N;

    // Per-thread fixed staging assignment (all 16B transfers):
    //  A tile: 32x32 = 256 float4 -> 1 per thread
    //  B tile: 32x128 = 1024 float4 -> 4 per thread
    const int ar  = tid >> 3, ac = (tid & 7) << 2;

    auto stage = [&](int bufsel, int k0) {
        async_copy16(&A[(size_t)(m0 + ar) * K + (k0 + ac)], &sA[bufsel][ar * LDA + ac]);
        #pragma unroll
        for (int j = 0; j < 4; ++j) {
            const int idx = tid + j * 256;
            const int br = idx >> 5, bc = (idx & 31) << 2;
            async_copy16(&B[(size_t)(k0 + br) * N + (n0 + bc)], &sB[bufsel][br * LDB + bc]);
        }
    };

    v8f acc0 = {0.f, 0.f, 0.f, 0.f, 0.f, 0.f, 0.f, 0.f};
    v8f acc1 = {0.f, 0.f, 0.f, 0.f, 0.f, 0.f, 0.f, 0.f};

    // prologue: stage tile 0 into buffer 0
    stage(0, 0);
    wait_async_all();
    __syncthreads();

    int buf = 0;
    for (int k0 = 0; k0 < K; k0 += TILE_K) {
        const int nbuf = buf ^ 1;
        if (k0 + TILE_K < K) stage(nbuf, k0 + TILE_K);   // overlap with compute

        #pragma unroll
        for (int kk = 0; kk < TILE_K; kk += 4) {
            v2f a, b0, b1;
            // A 16x4 fragment: VGPR v holds K = kk + 2*half + v, row M = lrow
            a.x = sA[buf][(wm0 + lrow) * LDA + kk + 2 * half + 0];
            a.y = sA[buf][(wm0 + lrow) * LDA + kk + 2 * half + 1];
            // B 4x16 fragments for the two N sub-tiles
            b0.x = sB[buf][(kk + 2 * half + 0) * LDB + wn0 + lrow];
            b0.y = sB[buf][(kk + 2 * half + 1) * LDB + wn0 + lrow];
            b1.x = sB[buf][(kk + 2 * half + 0) * LDB + wn0 + 16 + lrow];
            b1.y = sB[buf][(kk + 2 * half + 1) * LDB + wn0 + 16 + lrow];
            acc0 = __builtin_amdgcn_wmma_f32_16x16x4_f32(
                false, a, false, b0, (short)0, acc0, false, false);
            acc1 = __builtin_amdgcn_wmma_f32_16x16x4_f32(
                false, a, false, b1, (short)0, acc1, false, false);
        }

        wait_async_all();     // our async writes into nbuf have landed
        __syncthreads();      // everyone done reading buf / writing nbuf
        buf = nbuf;
    }

    // C/D layout: VGPR v -> row (v + 8*half), col lrow
    const int col0 = n0 + wn0 + lrow;
    const int col1 = col0 + 16;
    const float bv0 = bias ? bias[col0] : 0.0f;
    const float bv1 = bias ? bias[col1] : 0.0f;
    #pragma unroll
    for (int v = 0; v < 8; ++v) {
        const int row = m0 + wm0 + v + 8 * half;
        C[(size_t)row * N + col0] = acc0[v] + bv0;
        C[(size_t)row * N + col1] = acc1[v] + bv1;
    }
}

// One block per query; wave = head, lane = neighbor (K_NEIGH == warpSize == 32).
__global__ __launch_bounds__(256)
void sparse_attn(const float* __restrict__ Q,        // (B*HW, 512)
                 const float* __restrict__ KV,       // (B*L, 1024) = [k(512) | v(512)]
                 const long long* __restrict__ nidx, // (B*HW, 32) int64
                 const float* __restrict__ bias,     // (B*HW, 32)
                 float* __restrict__ O,              // (B*HW, 512)
                 int HW, int L) {
    __shared__ __align__(16) float sq[ATT_INNER];
    __shared__ float sattn[ATT_HEADS][ATT_KN];
    __shared__ int   sidx[ATT_KN];

    const int tid  = threadIdx.x;
    const int qid  = blockIdx.x;         // 0 .. B*HW-1
    const int b    = qid / HW;
    const int h    = tid >> 5;           // head
    const int lane = tid & 31;           // neighbor

    // stage q row (16B vector) + neighbor indices
    *reinterpret_cast<v4f*>(&sq[tid * 2]) =
        *reinterpret_cast<const v4f*>(&Q[(size_t)qid * ATT_INNER + tid * 2]);
    if (tid < ATT_KN) sidx[tid] = (int)nidx[(size_t)qid * ATT_KN + tid];
    __syncthreads();

    // sim[h][lane] = scale * <q_h, K[idx_lane]_h> + bias  (16B-vectorized dot)
    const int kb = sidx[lane];
    const v4f* __restrict__ k4 =
        reinterpret_cast<const v4f*>(&KV[((size_t)b * L + kb) * 1024 + h * ATT_DH]);
    const v4f* __restrict__ q4 =
        reinterpret_cast<const v4f*>(&sq[h * ATT_DH]);
    float sim = 0.0f;
    #pragma unroll
    for (int d4 = 0; d4 < ATT_DH / 4; ++d4) {
        const v4f kk = k4[d4];
        const v4f qq = q4[d4];
        sim += qq.x * kk.x + qq.y * kk.y + qq.z * kk.z + qq.w * kk.w;
    }
    sim = sim * ATT_SCALE + bias[(size_t)qid * ATT_KN + lane];

    // wave32 softmax across the 32 neighbors
    float mx = sim;
    #pragma unroll
    for (int off = 16; off > 0; off >>= 1) mx = fmaxf(mx, __shfl_xor(mx, off, 32));
    const float p = __expf(sim - mx);
    float s = p;
    #pragma unroll
    for (int off = 16; off > 0; off >>= 1) s += __shfl_xor(s, off, 32);
    sattn[h][lane] = p / s;
    __syncthreads();

    // out[h][d] = sum_k attn[k] * V[idx_k][h][d]; lane now covers d = lane, lane+32
    float o0 = 0.0f, o1 = 0.0f;
    #pragma unroll 4
    for (int k = 0; k < ATT_KN; ++k) {
        const float a = sattn[h][k];
        const float* __restrict__ vrow =
            &KV[((size_t)b * L + sidx[k]) * 1024 + 512 + h * ATT_DH];
        o0 += a * vrow[lane];
        o1 += a * vrow[lane + 32];
    }
    O[(size_t)qid * ATT_INNER + h * ATT_DH + lane]      = o0;
    O[(size_t)qid * ATT_INNER + h * ATT_DH + lane + 32] = o1;
}

extern "C" void kernel_launch(void* const* d_in, const int* in_sizes, int n_in,
                              void* d_out, int out_size, void* d_ws, size_t ws_size,
                              hipStream_t stream) {
    const float*     x    = (const float*)d_in[0];       // (B, HW, 512)
    const float*     ctx  = (const float*)d_in[1];       // (B, L, 512)
    const long long* nidx = (const long long*)d_in[2];   // (B, HW, 32) int64
    const float*     bias = (const float*)d_in[3];       // (B, HW, 32)
    const float*     Wq   = (const float*)d_in[4];       // (512, 512)
    const float*     Wkv  = (const float*)d_in[5];       // (512, 1024)
    const float*     Wo   = (const float*)d_in[6];       // (512, 512)
    const float*     bo   = (const float*)d_in[7];       // (512,)
    float*           out  = (float*)d_out;               // (B, HW, 512)

    const int Bq = 2, HW = 8192, D = 512, L = 4096;
    const int MQ = Bq * HW;   // 16384 query rows
    const int MK = Bq * L;    // 8192 context rows

    // workspace: q (32MB) | kv (32MB) | out_inner (32MB)
    float* q  = (float*)d_ws;
    float* kv = q  + (size_t)MQ * D;
    float* oi = kv + (size_t)MK * (2 * D);

    const dim3 blk(256);

    // q = x @ Wq                (16384 x 512 x 512)
    gemm_wmma_f32<<<dim3(D / TILE_N, MQ / TILE_M), blk, 0, stream>>>(
        x, Wq, nullptr, q, MQ, D, D);
    // kv = ctx @ Wkv            (8192 x 1024 x 512)
    gemm_wmma_f32<<<dim3((2 * D) / TILE_N, MK / TILE_M), blk, 0, stream>>>(
        ctx, Wkv, nullptr, kv, MK, 2 * D, D);
    // gathered attention (kv stays L2-resident: 32MB < 192MB)
    sparse_attn<<<dim3(MQ), blk, 0, stream>>>(q, kv, nidx, bias, oi, HW, L);
    // out = oi @ Wo + bo        (16384 x 512 x 512)
    gemm_wmma_f32<<<dim3(D / TILE_N, MQ / TILE_M), blk, 0, stream>>>(
        oi, Wo, bo, out, MQ, D, D);
}